// MagnitudeMaxPool2D_64991445123082
// MI455X (gfx1250) — compile-verified
//
#include <hip/hip_runtime.h>

// Magnitude-argmax 2x2 pooling on complex (re,im) data.
// Input  x:   [B, 512, 512, 32, 2] f32  (innermost (re,im) pairs)
// Output out: [B, 256, 256, 32, 2] f32
//
// Pure streaming kernel: each thread owns one (b, ho, wo, channel-pair),
// issues 4x global_load_b128 (NT hint) + 1x global_store_b128 (NT hint).

typedef __attribute__((ext_vector_type(4))) float v4f;

__global__ __launch_bounds__(256) void magmaxpool2x2_kernel(
    const float* __restrict__ x, float* __restrict__ out, int total)
{
    int tid = blockIdx.x * blockDim.x + threadIdx.x;
    if (tid >= total) return;

    // Fixed shape constants for [B,512,512,32,2], pool=2.
    constexpr int C2_SHIFT = 4;   // 16 channel-pairs per pixel (32 ch / 2)
    constexpr int WO_SHIFT = 8;   // Wo = 256
    constexpr int HO_SHIFT = 8;   // Ho = 256
    constexpr int PIX  = 64;      // floats per pixel (32*2)
    constexpr int ROW  = 512 * PIX;   // floats per input row
    constexpr int OROW = 256 * PIX;   // floats per output row

    const int c2 = tid & 15;
    int t  = tid >> C2_SHIFT;
    const int wo = t & 255;
    t >>= WO_SHIFT;
    const int ho = t & 255;
    const int b  = t >> HO_SHIFT;

    const size_t in_base =
        (size_t)(b * 512 + 2 * ho) * (size_t)ROW + (size_t)(2 * wo) * PIX + c2 * 4;

    const v4f* p00 = (const v4f*)(x + in_base);
    const v4f* p01 = (const v4f*)(x + in_base + PIX);
    const v4f* p10 = (const v4f*)(x + in_base + ROW);
    const v4f* p11 = (const v4f*)(x + in_base + ROW + PIX);

    // Non-temporal 128-bit loads: stream through L2 with TH_NT.
    v4f a = __builtin_nontemporal_load(p00);
    v4f c = __builtin_nontemporal_load(p01);
    v4f d = __builtin_nontemporal_load(p10);
    v4f e = __builtin_nontemporal_load(p11);

    v4f r;

    // Channel 0 of the pair: lanes .x = re, .y = im
    {
        float m0 = a.x * a.x + a.y * a.y;
        float m1 = c.x * c.x + c.y * c.y;
        float m2 = d.x * d.x + d.y * d.y;
        float m3 = e.x * e.x + e.y * e.y;
        float best = m0, br = a.x, bi = a.y;
        if (m1 > best) { best = m1; br = c.x; bi = c.y; }
        if (m2 > best) { best = m2; br = d.x; bi = d.y; }
        if (m3 > best) {            br = e.x; bi = e.y; }
        r.x = br; r.y = bi;
    }
    // Channel 1 of the pair: lanes .z = re, .w = im
    {
        float m0 = a.z * a.z + a.w * a.w;
        float m1 = c.z * c.z + c.w * c.w;
        float m2 = d.z * d.z + d.w * d.w;
        float m3 = e.z * e.z + e.w * e.w;
        float best = m0, br = a.z, bi = a.w;
        if (m1 > best) { best = m1; br = c.z; bi = c.w; }
        if (m2 > best) { best = m2; br = d.z; bi = d.w; }
        if (m3 > best) {            br = e.z; bi = e.w; }
        r.z = br; r.w = bi;
    }

    const size_t out_base =
        (size_t)(b * 256 + ho) * (size_t)OROW + (size_t)wo * PIX + c2 * 4;
    __builtin_nontemporal_store(r, (v4f*)(out + out_base));
}

extern "C" void kernel_launch(void* const* d_in, const int* in_sizes, int n_in,
                              void* d_out, int out_size, void* d_ws, size_t ws_size,
                              hipStream_t stream) {
    (void)n_in; (void)d_ws; (void)ws_size; (void)in_sizes;
    const float* x = (const float*)d_in[0];
    float* out = (float*)d_out;

    // One thread per output channel-pair (float4): total = out_size / 4.
    int total = out_size / 4;              // 4*256*256*32*2 / 4 = 4,194,304
    int block = 256;
    int grid  = (total + block - 1) / block;
    magmaxpool2x2_kernel<<<grid, block, 0, stream>>>(x, out, total);
}